// NodeNet_26053271618006
// MI455X (gfx1250) — compile-verified
//
#include <hip/hip_runtime.h>
#include <hip/hip_bf16.h>
#include <math.h>

typedef float v2f __attribute__((ext_vector_type(2)));
typedef float v8f __attribute__((ext_vector_type(8)));

struct Gate { unsigned char op, b0, b1, cs; };
// op 0: RY on bit b0 with cos/sin table index cs
// op 1: CX, control bit b0, target bit b1
#define G_RY(b,c) {0,(unsigned char)(b),0,(unsigned char)(c)}
#define G_CX(c,t) {1,(unsigned char)(c),(unsigned char)(t),0}

// cs index: 0..23 = encoding angles M[n][q]; 24+t = theta[t]

// ---- gates with support in qubits 0..11 (state "v", bit = qubit) ----
__constant__ Gate V_GATES[42] = {
  G_RY(0,0),G_RY(1,1),G_RY(2,2),G_RY(3,3),G_RY(4,4),G_RY(5,5),
  G_RY(6,6),G_RY(7,7),G_RY(8,8),G_RY(9,9),G_RY(10,10),G_RY(11,11),
  G_RY(0,24),G_RY(1,25),G_CX(0,1),
  G_RY(2,26),G_RY(3,27),G_CX(3,2),
  G_RY(4,28),G_RY(5,29),G_CX(4,5),
  G_RY(6,30),G_RY(7,31),G_CX(7,6),
  G_RY(8,32),G_RY(9,33),G_CX(8,9),
  G_RY(10,34),G_RY(11,35),G_CX(11,10),
  G_RY(1,48),G_RY(2,49),G_CX(1,2),
  G_RY(5,50),G_RY(6,51),G_CX(6,5),
  G_RY(9,52),G_RY(10,53),G_CX(9,10),
  G_RY(2,60),G_RY(5,61),G_CX(2,5)
};

// ---- gates with support in qubits 12..23 (state "u", bit = qubit-12) ----
__constant__ Gate U_GATES[42] = {
  G_RY(0,12),G_RY(1,13),G_RY(2,14),G_RY(3,15),G_RY(4,16),G_RY(5,17),
  G_RY(6,18),G_RY(7,19),G_RY(8,20),G_RY(9,21),G_RY(10,22),G_RY(11,23),
  G_RY(0,36),G_RY(1,37),G_CX(0,1),
  G_RY(2,38),G_RY(3,39),G_CX(3,2),
  G_RY(4,40),G_RY(5,41),G_CX(4,5),
  G_RY(6,42),G_RY(7,43),G_CX(7,6),
  G_RY(8,44),G_RY(9,45),G_CX(8,9),
  G_RY(10,46),G_RY(11,47),G_CX(11,10),
  G_RY(1,54),G_RY(2,55),G_CX(2,1),
  G_RY(5,56),G_RY(6,57),G_CX(5,6),
  G_RY(9,58),G_RY(10,59),G_CX(10,9),
  G_RY(6,64),G_RY(9,65),G_CX(9,6)
};

// ---- tail gates on meas-space bits: q2->0, q5->1, q10->2, q13->3, q18->4, q21->5 ----
__constant__ Gate W_GATES[23] = {
  G_RY(2,62),G_RY(3,63),G_CX(3,2),   // ry38@10 ry39@13 cx(13,10)
  G_RY(1,66),G_RY(2,67),G_CX(1,2),   // ry42@5  ry43@10 cx(5,10)
  G_RY(2,68),G_RY(4,69),G_CX(2,4),   // ry44@10 ry45@18 cx(10,18)
  G_RY(1,70),G_RY(2,71),G_CX(2,1),   // ry46@5  ry47@10 cx(10,5)
  G_RY(4,72),G_RY(5,73),G_CX(4,5),   // ry48@18 ry49@21 cx(18,21)
  G_RY(0,74),G_RY(1,75),G_CX(1,0),   // ry50@2  ry51@5  cx(5,2)
  G_RY(0,76),G_RY(1,77),G_RY(2,78),G_RY(4,79),G_RY(5,80) // ry52..56
};

// deposit helpers: meas bits of v are {2,5,10}; complement {0,1,3,4,6,7,8,9,11}
__device__ __forceinline__ unsigned xoff_v(int x) {
  return ((x & 1) << 2) | (((x >> 1) & 1) << 5) | (((x >> 2) & 1) << 10);
}
__device__ __forceinline__ unsigned expand_v(int y) {
  return (y & 0x3) | ((y & 0xC) << 1) | ((y & 0xF0) << 2) | ((y & 0x100) << 3);
}
// meas bits of u are {1,6,9}; complement {0,2,3,4,5,7,8,10,11}
__device__ __forceinline__ unsigned xoff_u(int x) {
  return ((x & 1) << 1) | (((x >> 1) & 1) << 6) | (((x >> 2) & 1) << 9);
}
__device__ __forceinline__ unsigned expand_u(int y) {
  return (y & 0x1) | ((y & 0x1E) << 1) | ((y & 0x60) << 2) | ((y & 0x180) << 3);
}

// run a 12-qubit (4096-amplitude) statevector through a gate list, in LDS
__device__ void run_state_4096(float* st, const Gate* gates, int ng,
                               const float2* cs, int tid) {
  for (int i = tid; i < 4096; i += 256) st[i] = (i == 0) ? 1.f : 0.f;
  __syncthreads();
  for (int g = 0; g < ng; ++g) {
    const Gate gd = gates[g];
    if (gd.op == 0) {
      const float c = cs[gd.cs].x, s = cs[gd.cs].y;
      const unsigned b = gd.b0, bm = 1u << b;
      for (unsigned p = tid; p < 2048u; p += 256u) {
        const unsigned i0 = ((p >> b) << (b + 1)) | (p & (bm - 1u));
        const unsigned i1 = i0 | bm;
        const float a0 = st[i0], a1 = st[i1];
        st[i0] = c * a0 - s * a1;
        st[i1] = s * a0 + c * a1;
      }
    } else {
      const unsigned bc = gd.b0, bt = gd.b1;
      const unsigned mt = 1u << bt, mc = 1u << bc;
      for (unsigned p = tid; p < 2048u; p += 256u) {
        const unsigned i0 = ((p >> bt) << (bt + 1)) | (p & (mt - 1u));
        if (i0 & mc) {
          const unsigned i1 = i0 | mt;
          const float a0 = st[i0];
          st[i0] = st[i1];
          st[i1] = a0;
        }
      }
    }
    __syncthreads();
  }
}

__launch_bounds__(256)
__global__ void qgnn_node_kernel(const float* __restrict__ X,
                                 const float* __restrict__ ew,
                                 const float* __restrict__ Ri,
                                 const float* __restrict__ Ro,
                                 const float* __restrict__ theta,
                                 float* __restrict__ out) {
  __shared__ float  buf[4096];    // v -> u -> T = U*rho
  __shared__ float  Umat[4096];   // 64x64 tail unitary
  __shared__ float  rho[4096];    // 64x64 reduced density (Rv x Ru)
  __shared__ float2 cs[81];
  __shared__ float  Mang[24];
  __shared__ float  Rv[64], Ru[64];
  __shared__ float  dvec[64];

  const int tid = threadIdx.x;
  const int n   = blockIdx.x;          // node 0..3

  // ---- phase 0: classical GNN preprocessing -> 24 encoding angles ----
  if (tid < 24) {
    float ang;
    if (tid < 16) {
      const int  d    = tid & 7;
      const bool isMi = (tid < 8);
      float acc = 0.f;
      for (int e = 0; e < 16; ++e) {
        float b = 0.f;
        for (int k = 0; k < 4; ++k)
          b += (isMi ? Ro[k * 16 + e] : Ri[k * 16 + e]) * X[k * 8 + d];
        acc += (isMi ? Ri[n * 16 + e] : Ro[n * 16 + e]) * ew[e] * b;
      }
      ang = acc;
    } else {
      ang = X[n * 8 + (tid - 16)];
    }
    Mang[tid] = ang;
  }
  __syncthreads();
  if (tid < 81) {
    const float a = 0.5f * ((tid < 24) ? Mang[tid] : theta[tid - 24]);
    cs[tid] = make_float2(cosf(a), sinf(a));
  }
  __syncthreads();

  // ---- phase 1: v over qubits 0..11, then Rv[x,x'] = sum_y v v ----
  run_state_4096(buf, V_GATES, 42, cs, tid);
  if (tid < 64) {
    const unsigned o1 = xoff_v(tid >> 3), o2 = xoff_v(tid & 7);
    float s = 0.f;
    for (int y = 0; y < 512; ++y) {
      const unsigned ey = expand_v(y);
      s += buf[o1 | ey] * buf[o2 | ey];
    }
    Rv[tid] = s;
  }
  __syncthreads();

  // ---- phase 2: u over qubits 12..23, then Ru ----
  run_state_4096(buf, U_GATES, 42, cs, tid);
  if (tid < 64) {
    const unsigned o1 = xoff_u(tid >> 3), o2 = xoff_u(tid & 7);
    float s = 0.f;
    for (int y = 0; y < 512; ++y) {
      const unsigned ey = expand_u(y);
      s += buf[o1 | ey] * buf[o2 | ey];
    }
    Ru[tid] = s;
  }
  __syncthreads();

  // ---- phase 3: build 64x64 tail unitary U (columns = evolved basis states) ----
  for (int i = tid; i < 4096; i += 256) Umat[i] = ((i >> 6) == (i & 63)) ? 1.f : 0.f;
  __syncthreads();
  for (int g = 0; g < 23; ++g) {
    const Gate gd = W_GATES[g];
    if (gd.op == 0) {
      const float c = cs[gd.cs].x, s = cs[gd.cs].y;
      const unsigned b = gd.b0, bm = 1u << b;
      for (int p = tid; p < 2048; p += 256) {
        const int col = p & 63, q = p >> 6;
        const unsigned r0 = (((unsigned)q >> b) << (b + 1)) | ((unsigned)q & (bm - 1u));
        const unsigned r1 = r0 | bm;
        const float a0 = Umat[r0 * 64 + col], a1 = Umat[r1 * 64 + col];
        Umat[r0 * 64 + col] = c * a0 - s * a1;
        Umat[r1 * 64 + col] = s * a0 + c * a1;
      }
    } else {
      const unsigned bc = gd.b0, bt = gd.b1;
      const unsigned mt = 1u << bt, mc = 1u << bc;
      for (int p = tid; p < 2048; p += 256) {
        const int col = p & 63, q = p >> 6;
        const unsigned r0 = (((unsigned)q >> bt) << (bt + 1)) | ((unsigned)q & (mt - 1u));
        if (r0 & mc) {
          const unsigned r1 = r0 | mt;
          const float a0 = Umat[r0 * 64 + col];
          Umat[r0 * 64 + col] = Umat[r1 * 64 + col];
          Umat[r1 * 64 + col] = a0;
        }
      }
    }
    __syncthreads();
  }

  // ---- phase 4: rho = Rv (bits 0..2) tensor Ru (bits 3..5) ----
  for (int i = tid; i < 4096; i += 256) {
    const int a = i >> 6, a2 = i & 63;
    rho[i] = Rv[(a & 7) * 8 + (a2 & 7)] * Ru[(a >> 3) * 8 + (a2 >> 3)];
  }
  __syncthreads();

  // ---- phase 5: T = U * rho via V_WMMA_F32_16X16X4_F32 (exact FP32) ----
  {
    const int lane = tid & 31;
    const int wave = tid >> 5;   // 8 wave32s
    const int half = lane >> 4;  // 0: K=0,1 ; 1: K=2,3 per 4-wide K step
    const int l16  = lane & 15;
    for (int it = 0; it < 2; ++it) {      // 16 tiles / 8 waves = 2 each
      const int tile = wave * 2 + it;
      const int ti = tile >> 2, tj = tile & 3;
      v8f acc = {};
      for (int kb = 0; kb < 16; ++kb) {
        const int k0 = kb * 4 + half * 2;
        v2f a, b;
        a.x = Umat[(ti * 16 + l16) * 64 + k0];
        a.y = Umat[(ti * 16 + l16) * 64 + k0 + 1];
        b.x = rho[(k0    ) * 64 + tj * 16 + l16];
        b.y = rho[(k0 + 1) * 64 + tj * 16 + l16];
        acc = __builtin_amdgcn_wmma_f32_16x16x4_f32(
            false, a, false, b, (short)0, acc, false, false);
      }
      for (int r = 0; r < 8; ++r) {
        const int row = ti * 16 + r + half * 8;   // C layout: lanes16-31 hold M=r+8
        buf[row * 64 + tj * 16 + l16] = acc[r];
      }
    }
  }
  __syncthreads();

  // ---- phase 6: d[a] = (U rho U^T)[a,a] = sum_k T[a,k] * U[a,k] ----
  if (tid < 64) {
    float s = 0.f;
    for (int k = 0; k < 64; ++k) s += buf[tid * 64 + k] * Umat[tid * 64 + k];
    dvec[tid] = s;
  }
  __syncthreads();

  // ---- phase 7: expectation values and output ----
  if (tid < 5) {
    const int mb[5] = {0, 1, 2, 4, 5};  // meas qubits 2,5,10,18,21 in meas-space bits
    const int b = mb[tid];
    float ev = 0.f;
    for (int a = 0; a < 64; ++a)
      ev += dvec[a] * (((a >> b) & 1) ? -1.f : 1.f);
    out[n * 5 + tid] = 3.14159265358979323846f * (1.f - ev);
  }
}

extern "C" void kernel_launch(void* const* d_in, const int* in_sizes, int n_in,
                              void* d_out, int out_size, void* d_ws, size_t ws_size,
                              hipStream_t stream) {
  (void)in_sizes; (void)n_in; (void)out_size; (void)d_ws; (void)ws_size;
  const float* X     = (const float*)d_in[0];
  const float* e     = (const float*)d_in[1];
  const float* Ri    = (const float*)d_in[2];
  const float* Ro    = (const float*)d_in[3];
  const float* theta = (const float*)d_in[4];
  qgnn_node_kernel<<<dim3(4), dim3(256), 0, stream>>>(
      X, e, Ri, Ro, theta, (float*)d_out);
}